// TopologyGCN_40587440947893
// MI455X (gfx1250) — compile-verified
//
#include <hip/hip_runtime.h>

// GCN, 2 layers, fp32. N=50000, D=H=128, E=800000.
// ws layout (floats): deg[N] | dis[N] | h1[N*H] | agg1[N*H] | h2[N*H]  (~77 MB)

typedef __attribute__((ext_vector_type(2))) float v2f;
typedef __attribute__((ext_vector_type(8))) float v8f;

#define FEAT 128            // D == H == 128
#define WAVES_PER_BLOCK 8   // 256 threads / wave32

// ---------------- utility ----------------
__global__ void zero_kernel(float* __restrict__ p, long n) {
    long i = (long)blockIdx.x * blockDim.x + threadIdx.x;
    long stride = (long)gridDim.x * blockDim.x;
    for (; i < n; i += stride) p[i] = 0.0f;
}

__global__ void degree_kernel(const long long* __restrict__ col,
                              float* __restrict__ deg, int E) {
    int e = blockIdx.x * blockDim.x + threadIdx.x;
    if (e < E) unsafeAtomicAdd(&deg[(long)col[e]], 1.0f);  // native global_atomic_add_f32
}

__global__ void rsqrt_kernel(const float* __restrict__ deg,
                             float* __restrict__ dis, int N) {
    int i = blockIdx.x * blockDim.x + threadIdx.x;
    if (i < N) {
        // +1.0f accounts for the self loop; deg >= 1 always, matching the
        // reference's where(deg>0, rsqrt(max(deg,1)), 0) exactly.
        dis[i] = rsqrtf(deg[i] + 1.0f);
    }
}

// ---------------- dense transform: Out = A[nrows x 128] @ W[128 x 128] ----------------
// One 16x16 output tile per wave via V_WMMA_F32_16X16X4_F32 (fp32-exact).
// The 128x16 slice of W for this block's n-tile is staged in LDS.
__global__ void __launch_bounds__(256)
gemm_wmma_kernel(const float* __restrict__ A, const float* __restrict__ W,
                 float* __restrict__ Out, int nrows) {
    __shared__ float Ws[FEAT * 16];                 // 8 KB: W[k][n0..n0+15]
    const int n0 = blockIdx.y * 16;
    for (int i = threadIdx.x; i < FEAT * 16; i += blockDim.x) {
        int k = i >> 4, j = i & 15;
        Ws[i] = W[k * FEAT + n0 + j];
    }
    __syncthreads();

    const int wave   = threadIdx.x >> 5;
    const int lane   = threadIdx.x & 31;
    const int mtiles = nrows >> 4;                  // 50000/16 = 3125 exact
    const int mtile  = blockIdx.x * WAVES_PER_BLOCK + wave;
    if (mtile >= mtiles) return;                    // wave-uniform: EXEC stays full

    const int half = lane >> 4;                     // 0: lanes 0-15, 1: lanes 16-31
    const int l16  = lane & 15;
    const int m    = mtile * 16 + l16;              // A-matrix row held by this lane

    v8f acc = {};
    const float* arow = A + (long)m * FEAT;
    #pragma unroll 4
    for (int k = 0; k < FEAT; k += 4) {
        // A 16x4 f32 frag: VGPR0 = K{0|2}, VGPR1 = K{1|3} per lane half -> contiguous pair
        v2f a = *(const v2f*)(arow + k + 2 * half);
        // B 4x16 f32 frag: lane holds column n=l16; VGPRs hold K{0,1} (half=0) / K{2,3} (half=1)
        v2f b;
        b.x = Ws[(k + 2 * half) * 16 + l16];
        b.y = Ws[(k + 2 * half + 1) * 16 + l16];
        acc = __builtin_amdgcn_wmma_f32_16x16x4_f32(
            /*neg_a=*/false, a, /*neg_b=*/false, b,
            /*c_mod=*/(short)0, acc, /*reuse_a=*/false, /*reuse_b=*/false);
    }

    // C/D layout: VGPR r holds row M = r + 8*half, column N = l16
    float* orow = Out + (long)mtile * 16 * FEAT + n0 + l16;
    #pragma unroll
    for (int r = 0; r < 8; ++r)
        orow[(long)(r + 8 * half) * FEAT] = acc[r];
}

// ---------------- edge scatter: out[col] += h[row] * dis[row]*dis[col] ----------------
// One wave per edge; each lane moves one float4 => one coalesced 512B row per wave.
__global__ void __launch_bounds__(256)
scatter_kernel(const float* __restrict__ h,
               const long long* __restrict__ row, const long long* __restrict__ col,
               const float* __restrict__ dis, float* __restrict__ out,
               int E, int N) {
    const int lane = threadIdx.x & 31;
    const int edge = blockIdx.x * (blockDim.x >> 5) + (threadIdx.x >> 5);
    const int total = E + N;                        // real edges + self loops
    if (edge >= total) return;

    long r, c;
    if (edge < E) { r = (long)row[edge]; c = (long)col[edge]; }
    else          { r = c = (long)(edge - E); }     // self loop

    const float* src = h + r * FEAT + lane * 4;
    __builtin_prefetch(src, 0, 0);                  // global_prefetch_b8
    const float norm = dis[r] * dis[c];
    const float4 v = *(const float4*)src;

    float* dst = out + c * FEAT + lane * 4;
    unsafeAtomicAdd(dst + 0, v.x * norm);
    unsafeAtomicAdd(dst + 1, v.y * norm);
    unsafeAtomicAdd(dst + 2, v.z * norm);
    unsafeAtomicAdd(dst + 3, v.w * norm);
}

// ---------------- epilogues ----------------
__global__ void bias_relu_kernel(float* __restrict__ a, const float* __restrict__ b, long n) {
    long i = (long)blockIdx.x * blockDim.x + threadIdx.x;
    if (i < n) {
        float v = a[i] + b[i & (FEAT - 1)];
        a[i] = v > 0.0f ? v : 0.0f;
    }
}

__global__ void bias_add_kernel(float* __restrict__ a, const float* __restrict__ b, long n) {
    long i = (long)blockIdx.x * blockDim.x + threadIdx.x;
    if (i < n) a[i] += b[i & (FEAT - 1)];
}

// ---------------- driver ----------------
extern "C" void kernel_launch(void* const* d_in, const int* in_sizes, int n_in,
                              void* d_out, int out_size, void* d_ws, size_t ws_size,
                              hipStream_t stream) {
    const float*     x   = (const float*)d_in[0];
    const float*     W1  = (const float*)d_in[1];
    const float*     b1  = (const float*)d_in[2];
    const float*     W2  = (const float*)d_in[3];
    const float*     b2  = (const float*)d_in[4];
    const long long* ei  = (const long long*)d_in[5];   // int64 edge_index [2, E]

    const int H = in_sizes[2];              // 128
    const int N = in_sizes[0] / H;          // D == H
    const int E = in_sizes[5] / 2;
    const long NH = (long)N * H;

    float* ws   = (float*)d_ws;
    float* deg  = ws;
    float* dis  = deg + N;
    float* h1   = dis + N;
    float* agg1 = h1 + NH;
    float* h2   = agg1 + NH;
    float* outp = (float*)d_out;

    const long long* rowp = ei;
    const long long* colp = ei + E;

    // zero accumulators (harness poisons out/ws with 0xAA)
    zero_kernel<<<1024, 256, 0, stream>>>(deg, (long)N);
    zero_kernel<<<4096, 256, 0, stream>>>(agg1, NH);
    zero_kernel<<<4096, 256, 0, stream>>>(outp, NH);

    degree_kernel<<<(E + 255) / 256, 256, 0, stream>>>(colp, deg, E);
    rsqrt_kernel<<<(N + 255) / 256, 256, 0, stream>>>(deg, dis, N);

    const int mtiles = N / 16;
    dim3 gg((mtiles + WAVES_PER_BLOCK - 1) / WAVES_PER_BLOCK, H / 16);
    const int totalEdges = E + N;
    const int sblocks = (totalEdges + WAVES_PER_BLOCK - 1) / WAVES_PER_BLOCK;
    const int nhblocks = (int)((NH + 255) / 256);

    // layer 1
    gemm_wmma_kernel<<<gg, 256, 0, stream>>>(x, W1, h1, N);
    scatter_kernel<<<sblocks, 256, 0, stream>>>(h1, rowp, colp, dis, agg1, E, N);
    bias_relu_kernel<<<nhblocks, 256, 0, stream>>>(agg1, b1, NH);

    // layer 2
    gemm_wmma_kernel<<<gg, 256, 0, stream>>>(agg1, W2, h2, N);
    scatter_kernel<<<sblocks, 256, 0, stream>>>(h2, rowp, colp, dis, outp, E, N);
    bias_add_kernel<<<nhblocks, 256, 0, stream>>>(outp, b2, NH);
}